// SGC_64836826300771
// MI455X (gfx1250) — compile-verified
//
#include <hip/hip_runtime.h>
#include <hip/hip_bf16.h>
#include <math.h>

typedef float v2f __attribute__((ext_vector_type(2)));
typedef float v8f __attribute__((ext_vector_type(8)));

#define D_FEAT 256
#define N_CLS  40

#if __has_builtin(__builtin_amdgcn_global_load_async_to_lds_b128)
#define HAVE_ASYNC_LDS 1
typedef int v4i_ __attribute__((ext_vector_type(4)));
typedef __attribute__((address_space(1))) v4i_* gas_ptr_t;   // global src (non-const int4*)
typedef __attribute__((address_space(3))) v4i_* las_ptr_t;   // LDS dst
#else
#define HAVE_ASYNC_LDS 0
#endif

__device__ __forceinline__ void wait_async0() {
#if __has_builtin(__builtin_amdgcn_s_wait_asynccnt)
  __builtin_amdgcn_s_wait_asynccnt(0);
#else
  asm volatile("s_wait_asynccnt 0x0" ::: "memory");
#endif
}

// ---------------------------------------------------------------------------
// 1) init counters
__global__ void k_init(int* __restrict__ cnt, int* __restrict__ cursor, int n) {
  int i = blockIdx.x * blockDim.x + threadIdx.x;
  if (i < n) { cnt[i] = 0; cursor[i] = 0; }
}

// 2) in-degree histogram over dst
__global__ void k_count(const int* __restrict__ ei, int* __restrict__ cnt,
                        long long E) {
  long long e = (long long)blockIdx.x * blockDim.x + threadIdx.x;
  if (e < E) {
    int d = ei[E + e];           // edge_index[1][e]
    atomicAdd(&cnt[d], 1);
  }
}

// 3) exclusive scan cnt -> rowptr (single block, chunked Hillis-Steele)
__launch_bounds__(1024)
__global__ void k_scan(const int* __restrict__ cnt, int* __restrict__ rowptr,
                       int n) {
  __shared__ int sdata[1024];
  __shared__ int s_carry;
  int tid = threadIdx.x;
  if (tid == 0) s_carry = 0;
  __syncthreads();
  for (int base = 0; base < n; base += 1024) {
    int i = base + tid;
    int v = (i < n) ? cnt[i] : 0;
    sdata[tid] = v;
    __syncthreads();
    for (int off = 1; off < 1024; off <<= 1) {
      int t = (tid >= off) ? sdata[tid - off] : 0;
      __syncthreads();
      sdata[tid] += t;
      __syncthreads();
    }
    if (i < n) rowptr[i] = s_carry + sdata[tid] - v;   // exclusive
    __syncthreads();
    if (tid == 1023) s_carry += sdata[1023];
    __syncthreads();
  }
  if (tid == 0) rowptr[n] = s_carry;
}

// 4) dinv = rsqrt(deg), deg = in-degree + 1 (self loop)
__global__ void k_dinv(const int* __restrict__ cnt, float* __restrict__ dinv,
                       int n) {
  int i = blockIdx.x * blockDim.x + threadIdx.x;
  if (i < n) dinv[i] = rsqrtf((float)(cnt[i] + 1));
}

// 5) counting-sort scatter: place src index of each edge into its dst segment
__global__ void k_scatter(const int* __restrict__ ei, const int* __restrict__ rowptr,
                          int* __restrict__ cursor, int* __restrict__ esrc,
                          long long E) {
  long long e = (long long)blockIdx.x * blockDim.x + threadIdx.x;
  if (e < E) {
    int s = ei[e];
    int d = ei[E + e];
    int pos = rowptr[d] + atomicAdd(&cursor[d], 1);
    esrc[pos] = s;
  }
}

// 6) SpMM propagate: O[n] = dinv[n]^2 * F[n] + sum_e dinv[src]*dinv[n]*F[src]
//    One wave per node, each lane owns 8 contiguous columns (2x float4).
//    F (102 MB) is L2-resident (192 MB L2); the gather loop is L2-latency
//    bound, so: scalarized edge index (readfirstlane) + global_prefetch of
//    the row 8 edges ahead.
__launch_bounds__(256)
__global__ void k_spmm(const float* __restrict__ F, float* __restrict__ O,
                       const int* __restrict__ rowptr, const int* __restrict__ esrc,
                       const float* __restrict__ dinv, int n) {
  int wave = (int)((blockIdx.x * blockDim.x + threadIdx.x) >> 5);
  int lane = threadIdx.x & 31;
  if (wave >= n) return;

  float dn = dinv[wave];
  const float* frow = F + (size_t)wave * D_FEAT + lane * 8;
  float4 f0 = *(const float4*)(frow);
  float4 f1 = *(const float4*)(frow + 4);
  float ws = dn * dn;  // self-loop norm
  float4 a0 = make_float4(ws * f0.x, ws * f0.y, ws * f0.z, ws * f0.w);
  float4 a1 = make_float4(ws * f1.x, ws * f1.y, ws * f1.z, ws * f1.w);

  int beg = rowptr[wave], end = rowptr[wave + 1];
  for (int p = beg; p < end; ++p) {
    int s = __builtin_amdgcn_readfirstlane(esrc[p]);   // wave-uniform -> SGPR
    int pf = p + 8;
    if (pf < end) {                                    // prefetch 8 edges ahead
      int sp = __builtin_amdgcn_readfirstlane(esrc[pf]);
      __builtin_prefetch(F + (size_t)sp * D_FEAT + lane * 8, 0, 3);
    }
    float wn = dinv[s] * dn;
    const float* sr = F + (size_t)s * D_FEAT + lane * 8;
    float4 g0 = *(const float4*)(sr);
    float4 g1 = *(const float4*)(sr + 4);
    a0.x = fmaf(wn, g0.x, a0.x); a0.y = fmaf(wn, g0.y, a0.y);
    a0.z = fmaf(wn, g0.z, a0.z); a0.w = fmaf(wn, g0.w, a0.w);
    a1.x = fmaf(wn, g1.x, a1.x); a1.y = fmaf(wn, g1.y, a1.y);
    a1.z = fmaf(wn, g1.z, a1.z); a1.w = fmaf(wn, g1.w, a1.w);
  }
  float* orow = O + (size_t)wave * D_FEAT + lane * 8;
  *(float4*)(orow)     = a0;
  *(float4*)(orow + 4) = a1;
}

// --- A-tile staging into LDS: async DMA (gfx1250 GLOBAL_LOAD_ASYNC_TO_LDS)
//     when available, VGPR round-trip fallback otherwise. ---
template <int NTHREADS>
__device__ __forceinline__ void stage_A_tile(float* As, const float* src,
                                             int row0, int n) {
  bool full = (row0 + 16 <= n);
#if HAVE_ASYNC_LDS
  if (full) {
    for (int c = threadIdx.x; c < 16 * (D_FEAT / 4); c += NTHREADS) {
      int r = c >> 6, cc = c & 63;           // 64 x 16B chunks per row
      float* g = const_cast<float*>(src) + (size_t)(row0 + r) * D_FEAT + cc * 4;
      __builtin_amdgcn_global_load_async_to_lds_b128(
          (gas_ptr_t)g, (las_ptr_t)(As + c * 4), 0, 0);
    }
    wait_async0();
    __syncthreads();
    return;
  }
#endif
  for (int i = threadIdx.x; i < 16 * D_FEAT; i += NTHREADS) {
    int r = i >> 8, cc = i & 255;
    int gr = row0 + r;
    As[i] = (full || gr < n) ? src[(size_t)gr * D_FEAT + cc] : 0.0f;
  }
  __syncthreads();
}

// 7) GEMM1 (in place): H = relu(P @ W1 + b1). WMMA f32 16x16x4.
//    In-place safe: block reads only its own 16 rows (from LDS post-barrier).
__launch_bounds__(256)
__global__ void k_gemm1(float* __restrict__ B1, const float* __restrict__ W1,
                        const float* __restrict__ b1, int n) {
  __shared__ float As[16 * D_FEAT];
  int row0 = blockIdx.x * 16;
  stage_A_tile<256>(As, B1, row0, n);

  int wave  = threadIdx.x >> 5;
  int lane  = threadIdx.x & 31;
  int mlane = lane & 15;
  int koff  = (lane >> 4) << 1;        // 0 for lanes 0-15, 2 for lanes 16-31
  int mbase = (lane >> 4) << 3;        // C/D: M = r + 8*(lane>=16)
  bool full = (row0 + 16 <= n);

  for (int t = 0; t < 2; ++t) {
    int colbase = (wave + t * 8) * 16;
    v8f c = {};
    for (int k0 = 0; k0 < D_FEAT; k0 += 4) {
      v2f a, b;
      a.x = As[mlane * D_FEAT + k0 + koff];
      a.y = As[mlane * D_FEAT + k0 + koff + 1];
      b.x = W1[(size_t)(k0 + koff)     * D_FEAT + colbase + mlane];
      b.y = W1[(size_t)(k0 + koff + 1) * D_FEAT + colbase + mlane];
      c = __builtin_amdgcn_wmma_f32_16x16x4_f32(false, a, false, b,
                                                (short)0, c, false, false);
    }
    int col = colbase + mlane;
    float bias = b1[col];
    if (full) {
      float* outp = B1 + (size_t)(row0 + mbase) * D_FEAT + col;
#pragma unroll
      for (int r = 0; r < 8; ++r)
        outp[(size_t)r * D_FEAT] = fmaxf(c[r] + bias, 0.0f);
    } else {
      for (int r = 0; r < 8; ++r) {
        int gr = row0 + mbase + r;
        if (gr < n) B1[(size_t)gr * D_FEAT + col] = fmaxf(c[r] + bias, 0.0f);
      }
    }
  }
}

// 8) GEMM2: logits = P2 @ W2 + b2 (C=40 padded to 48 -> 3 col tiles, 3 waves)
__launch_bounds__(96)
__global__ void k_gemm2(const float* __restrict__ B2, const float* __restrict__ W2,
                        const float* __restrict__ b2, float* __restrict__ logits,
                        int n) {
  __shared__ float As[16 * D_FEAT];
  int row0 = blockIdx.x * 16;
  stage_A_tile<96>(As, B2, row0, n);

  int wave  = threadIdx.x >> 5;
  int lane  = threadIdx.x & 31;
  int mlane = lane & 15;
  int koff  = (lane >> 4) << 1;
  int mbase = (lane >> 4) << 3;
  bool full = (row0 + 16 <= n);

  int colbase = wave * 16;
  int col = colbase + mlane;
  bool cok = (col < N_CLS);

  v8f c = {};
  for (int k0 = 0; k0 < D_FEAT; k0 += 4) {
    v2f a, b;
    a.x = As[mlane * D_FEAT + k0 + koff];
    a.y = As[mlane * D_FEAT + k0 + koff + 1];
    b.x = cok ? W2[(size_t)(k0 + koff)     * N_CLS + col] : 0.0f;
    b.y = cok ? W2[(size_t)(k0 + koff + 1) * N_CLS + col] : 0.0f;
    c = __builtin_amdgcn_wmma_f32_16x16x4_f32(false, a, false, b,
                                              (short)0, c, false, false);
  }
  float bias = cok ? b2[col] : 0.0f;
  if (full) {
    if (cok) {
      float* outp = logits + (size_t)(row0 + mbase) * N_CLS + col;
#pragma unroll
      for (int r = 0; r < 8; ++r) outp[(size_t)r * N_CLS] = c[r] + bias;
    }
  } else {
    for (int r = 0; r < 8; ++r) {
      int gr = row0 + mbase + r;
      if (cok && gr < n) logits[(size_t)gr * N_CLS + col] = c[r] + bias;
    }
  }
}

// 9) log_softmax over 40 classes, one wave per row (shuffle reductions)
__launch_bounds__(256)
__global__ void k_lsm(const float* __restrict__ logits, float* __restrict__ out,
                      int n) {
  int wave = (int)((blockIdx.x * blockDim.x + threadIdx.x) >> 5);
  int lane = threadIdx.x & 31;
  if (wave >= n) return;
  const float* row = logits + (size_t)wave * N_CLS;
  float v0 = row[lane];
  float v1 = (lane + 32 < N_CLS) ? row[lane + 32] : -INFINITY;
  float m = fmaxf(v0, v1);
  for (int off = 16; off; off >>= 1) m = fmaxf(m, __shfl_xor(m, off, 32));
  float s = __expf(v0 - m) + ((lane + 32 < N_CLS) ? __expf(v1 - m) : 0.0f);
  for (int off = 16; off; off >>= 1) s += __shfl_xor(s, off, 32);
  float ls = __logf(s);
  float* orow = out + (size_t)wave * N_CLS;
  orow[lane] = v0 - m - ls;
  if (lane + 32 < N_CLS) orow[lane + 32] = v1 - m - ls;
}

// ---------------------------------------------------------------------------
extern "C" void kernel_launch(void* const* d_in, const int* in_sizes, int n_in,
                              void* d_out, int out_size, void* d_ws, size_t ws_size,
                              hipStream_t stream) {
  const float* x   = (const float*)d_in[0];
  const int*   ei  = (const int*)d_in[1];     // [2][E] flattened
  const float* W1  = (const float*)d_in[2];
  const float* b1  = (const float*)d_in[3];
  const float* W2  = (const float*)d_in[4];
  const float* b2  = (const float*)d_in[5];
  float* out = (float*)d_out;

  const int       N = in_sizes[0] / D_FEAT;
  const long long E = (long long)in_sizes[1] / 2;

  // workspace carve-out (256B aligned)
  char* w = (char*)d_ws;
  auto alloc = [&](size_t bytes) -> void* {
    void* p = (void*)w;
    w += (bytes + 255) & ~(size_t)255;
    return p;
  };
  int*   cnt    = (int*)  alloc((size_t)N * 4);
  int*   rowptr = (int*)  alloc((size_t)(N + 1) * 4);
  int*   cursor = (int*)  alloc((size_t)N * 4);
  float* dinv   = (float*)alloc((size_t)N * 4);
  int*   esrc   = (int*)  alloc((size_t)E * 4);
  float* B1     = (float*)alloc((size_t)N * D_FEAT * 4);   // P then H (in place)
  float* B2     = (float*)alloc((size_t)N * D_FEAT * 4);   // P2
  float* logits = (float*)alloc((size_t)N * N_CLS * 4);

  int nbN = (N + 255) / 256;
  int nbE = (int)((E + 255) / 256);

  k_init   <<<nbN, 256, 0, stream>>>(cnt, cursor, N);
  k_count  <<<nbE, 256, 0, stream>>>(ei, cnt, E);
  k_scan   <<<1, 1024, 0, stream>>>(cnt, rowptr, N);
  k_dinv   <<<nbN, 256, 0, stream>>>(cnt, dinv, N);
  k_scatter<<<nbE, 256, 0, stream>>>(ei, rowptr, cursor, esrc, E);

  int nbW = (N + 7) / 8;     // 8 waves (nodes) per 256-thread block
  int nbT = (N + 15) / 16;   // 16-row GEMM blocks

  k_spmm <<<nbW, 256, 0, stream>>>(x, B1, rowptr, esrc, dinv, N);   // P = A_hat x
  k_gemm1<<<nbT, 256, 0, stream>>>(B1, W1, b1, N);                  // H = relu(PW1+b1)
  k_spmm <<<nbW, 256, 0, stream>>>(B1, B2, rowptr, esrc, dinv, N);  // P2 = A_hat H
  k_gemm2<<<nbT,  96, 0, stream>>>(B2, W2, b2, logits, N);          // logits
  k_lsm  <<<nbW, 256, 0, stream>>>(logits, out, N);                 // log_softmax
}